// TransformerConv_81003083203032
// MI455X (gfx1250) — compile-verified
//
#include <hip/hip_runtime.h>
#include <math.h>

typedef __attribute__((ext_vector_type(2))) float v2f;
typedef __attribute__((ext_vector_type(8))) float v8f;

#define EDIM 16

// Monotone float atomic-max via int/uint ordering trick.
__device__ __forceinline__ void atomic_max_f32(float* addr, float val) {
    if (val >= 0.0f) {
        atomicMax((int*)addr, __float_as_int(val));
    } else {
        atomicMin((unsigned int*)addr, (unsigned int)__float_as_int(val));
    }
}

__global__ void init_buffers(float* __restrict__ mbuf, float* __restrict__ denom,
                             float* __restrict__ out, int n4, int n128) {
    int i = blockIdx.x * blockDim.x + threadIdx.x;
    if (i < n128) out[i] = 0.0f;
    if (i < n4) { mbuf[i] = -3.402823466e38f; denom[i] = 0.0f; }
}

// Fused [q|k|v|xskip] = x @ [Wq|Wk|Wv|Wskip] + bias via V_WMMA_F32_16X16X4_F32.
// Block = 4 waves; all waves share one 16x128 x-tile in LDS (stride 132 to
// avoid 16-way bank conflicts on the column-pair reads). Each wave computes
// one 16x16 tile of the 512-wide fused output. Full-tile fast paths keep the
// prologue/epilogue branch-free for the (common) interior blocks.
__global__ __launch_bounds__(128) void fused_qkvs_gemm(
    const float* __restrict__ x,
    const float* __restrict__ Wq, const float* __restrict__ bq,
    const float* __restrict__ Wk, const float* __restrict__ bk,
    const float* __restrict__ Wv, const float* __restrict__ bv,
    const float* __restrict__ Wsk, const float* __restrict__ bsk,
    float* __restrict__ q, float* __restrict__ k,
    float* __restrict__ v, float* __restrict__ xsk, int N)
{
    __shared__ float xs[16 * 132];
    const int lane = threadIdx.x;
    const int wave = threadIdx.y;
    const int m0   = blockIdx.x * 16;
    const bool full = (m0 + 16 <= N);   // wave-uniform

    // cooperative load of x tile (16 rows x 128 cols)
    const int tid = wave * 32 + lane;
    if (full) {
        // 512 float4 chunks; a 4-aligned chunk never crosses a 128-float row
        const float4* xsrc = (const float4*)(x + (size_t)m0 * 128);
#pragma unroll
        for (int j = 0; j < 4; ++j) {
            const int i4 = tid + j * 128;
            const float4 val = xsrc[i4];
            const int r = i4 >> 5, c = (i4 & 31) * 4;
            float* dstp = xs + r * 132 + c;
            dstp[0] = val.x; dstp[1] = val.y; dstp[2] = val.z; dstp[3] = val.w;
        }
    } else {
        for (int i = tid; i < 16 * 128; i += 128) {
            int r = i >> 7, c = i & 127;
            int gr = m0 + r;
            xs[r * 132 + c] = (gr < N) ? x[(size_t)gr * 128 + c] : 0.0f;
        }
    }
    __syncthreads();

    const int ct  = blockIdx.y * 4 + wave;   // fused col tile 0..31
    const int mat = ct >> 3;                 // 0=q 1=k 2=v 3=skip
    const int n0  = (ct & 7) * 16;           // col offset within matrix

    const float* W; const float* bias; float* dst;
    if (mat == 0)      { W = Wq;  bias = bq;  dst = q;   }
    else if (mat == 1) { W = Wk;  bias = bk;  dst = k;   }
    else if (mat == 2) { W = Wv;  bias = bv;  dst = v;   }
    else               { W = Wsk; bias = bsk; dst = xsk; }

    const int ncol  = n0 + (lane & 15);  // B/C/D: lane -> column
    const int khalf = lane >> 4;         // A/B: lane-half -> K pair
    const int arow  = lane & 15;         // A: lane -> row

    const float bval = bias[ncol];
    v8f acc;
#pragma unroll
    for (int r = 0; r < 8; ++r) acc[r] = bval;

#pragma unroll 4
    for (int kb = 0; kb < 128; kb += 4) {
        const int k0 = kb + khalf * 2;
        v2f a;
        a.x = xs[arow * 132 + k0];
        a.y = xs[arow * 132 + k0 + 1];
        v2f b;
        b.x = W[(size_t)k0 * 128 + ncol];
        b.y = W[(size_t)(k0 + 1) * 128 + ncol];
        acc = __builtin_amdgcn_wmma_f32_16x16x4_f32(
            /*neg_a=*/false, a, /*neg_b=*/false, b,
            /*c_mod=*/(short)0, acc, /*reuse_a=*/false, /*reuse_b=*/false);
    }

    const int mbase = m0 + khalf * 8;    // C/D layout: VGPR r -> row r / r+8
    if (full) {
        float* dp = dst + (size_t)mbase * 128 + ncol;
#pragma unroll
        for (int r = 0; r < 8; ++r) dp[(size_t)r * 128] = acc[r];
    } else {
#pragma unroll
        for (int r = 0; r < 8; ++r) {
            int mr = mbase + r;
            if (mr < N) dst[(size_t)mr * 128 + ncol] = acc[r];
        }
    }
}

// One wave per edge. Lane owns 4 channels; head = lane>>3.
// alpha[e,h] = (q[row,h] . (k[col,h] + ea@We[h])) / sqrt(32); atomic-max into m.
__global__ __launch_bounds__(128) void edge_logits(
    const float* __restrict__ edge_attr, const int* __restrict__ ei,
    const float* __restrict__ We,
    const float* __restrict__ q, const float* __restrict__ k,
    float* __restrict__ alpha, float* __restrict__ mbuf, int E)
{
    const int e = blockIdx.x * 4 + threadIdx.y;
    if (e >= E) return;
    const int lane = threadIdx.x;
    const int row = ei[e];
    const int col = ei[(size_t)E + e];
    const int c0 = lane * 4;

    float ek0 = 0.f, ek1 = 0.f, ek2 = 0.f, ek3 = 0.f;
    const float* ea = edge_attr + (size_t)e * EDIM;
#pragma unroll
    for (int d = 0; d < EDIM; ++d) {
        const float a = ea[d];
        const float* w = We + d * 128 + c0;
        ek0 = fmaf(a, w[0], ek0);
        ek1 = fmaf(a, w[1], ek1);
        ek2 = fmaf(a, w[2], ek2);
        ek3 = fmaf(a, w[3], ek3);
    }
    const float* qp = q + (size_t)row * 128 + c0;
    const float* kp = k + (size_t)col * 128 + c0;
    float p = qp[0] * (kp[0] + ek0) + qp[1] * (kp[1] + ek1)
            + qp[2] * (kp[2] + ek2) + qp[3] * (kp[3] + ek3);
    // reduce the 8 lanes of each head
    p += __shfl_xor(p, 1);
    p += __shfl_xor(p, 2);
    p += __shfl_xor(p, 4);
    p *= 0.17677669529663687f;  // 1/sqrt(32)
    if ((lane & 7) == 0) {
        const int h = lane >> 3;
        alpha[(size_t)e * 4 + h] = p;
        atomic_max_f32(mbuf + (size_t)row * 4 + h, p);
    }
}

// One wave per edge: ex = exp(alpha - m[row]); denom[row,h] += ex;
// out[row,c] += ex * v[col,c]. Normalization is applied in finalize.
__global__ __launch_bounds__(128) void edge_scatter(
    const int* __restrict__ ei,
    const float* __restrict__ alpha, const float* __restrict__ mbuf,
    const float* __restrict__ v,
    float* __restrict__ denom, float* __restrict__ out, int E)
{
    const int e = blockIdx.x * 4 + threadIdx.y;
    if (e >= E) return;
    const int lane = threadIdx.x;
    const int row = ei[e];
    const int col = ei[(size_t)E + e];
    const int h = lane >> 3;

    const float a  = alpha[(size_t)e * 4 + h];
    const float mm = mbuf[(size_t)row * 4 + h];
    const float ex = expf(a - mm);
    if ((lane & 7) == 0) atomicAdd(denom + (size_t)row * 4 + h, ex);

    const int c0 = lane * 4;
    const float* vp = v + (size_t)col * 128 + c0;
    float* op = out + (size_t)row * 128 + c0;
    atomicAdd(op + 0, ex * vp[0]);
    atomicAdd(op + 1, ex * vp[1]);
    atomicAdd(op + 2, ex * vp[2]);
    atomicAdd(op + 3, ex * vp[3]);
}

// One wave per node: normalize, gated skip: beta = sigmoid([out|sk|out-sk]@Wb)
__global__ __launch_bounds__(128) void finalize(
    const float* __restrict__ denom, const float* __restrict__ xsk,
    const float* __restrict__ Wbeta, float* __restrict__ out, int N)
{
    const int n = blockIdx.x * 4 + threadIdx.y;
    if (n >= N) return;
    const int lane = threadIdx.x;
    const int c0 = lane * 4;
    const int h = lane >> 3;

    const float inv = 1.0f / (denom[(size_t)n * 4 + h] + 1e-16f);
    float* op = out + (size_t)n * 128 + c0;
    const float o0 = op[0] * inv, o1 = op[1] * inv, o2 = op[2] * inv, o3 = op[3] * inv;
    const float* sp = xsk + (size_t)n * 128 + c0;
    const float s0 = sp[0], s1 = sp[1], s2 = sp[2], s3 = sp[3];

    const float* w1 = Wbeta + c0;
    const float* w2 = Wbeta + 128 + c0;
    const float* w3 = Wbeta + 256 + c0;
    float p = o0 * w1[0] + o1 * w1[1] + o2 * w1[2] + o3 * w1[3]
            + s0 * w2[0] + s1 * w2[1] + s2 * w2[2] + s3 * w2[3]
            + (o0 - s0) * w3[0] + (o1 - s1) * w3[1]
            + (o2 - s2) * w3[2] + (o3 - s3) * w3[3];
    p += __shfl_xor(p, 1);
    p += __shfl_xor(p, 2);
    p += __shfl_xor(p, 4);
    p += __shfl_xor(p, 8);
    p += __shfl_xor(p, 16);
    const float beta = 1.0f / (1.0f + expf(-p));
    op[0] = beta * s0 + (1.0f - beta) * o0;
    op[1] = beta * s1 + (1.0f - beta) * o1;
    op[2] = beta * s2 + (1.0f - beta) * o2;
    op[3] = beta * s3 + (1.0f - beta) * o3;
}

extern "C" void kernel_launch(void* const* d_in, const int* in_sizes, int n_in,
                              void* d_out, int out_size, void* d_ws, size_t ws_size,
                              hipStream_t stream) {
    const float* x          = (const float*)d_in[0];
    const float* edge_attr  = (const float*)d_in[1];
    const int*   edge_index = (const int*)d_in[2];
    const float* Wq    = (const float*)d_in[3];
    const float* bq    = (const float*)d_in[4];
    const float* Wk    = (const float*)d_in[5];
    const float* bk    = (const float*)d_in[6];
    const float* Wv    = (const float*)d_in[7];
    const float* bv    = (const float*)d_in[8];
    const float* We    = (const float*)d_in[9];
    const float* Wsk   = (const float*)d_in[10];
    const float* bsk   = (const float*)d_in[11];
    const float* Wbeta = (const float*)d_in[12];
    float* out = (float*)d_out;

    const int N = in_sizes[0] / 128;
    const int E = in_sizes[2] / 2;

    // workspace layout (floats): q,k,v,xskip [N*128 each], m,denom [N*4], alpha [E*4]
    float* ws  = (float*)d_ws;
    const size_t nn = (size_t)N * 128;
    float* q     = ws;
    float* k     = q + nn;
    float* v     = k + nn;
    float* xsk   = v + nn;
    float* mbuf  = xsk + nn;
    float* denom = mbuf + (size_t)N * 4;
    float* alpha = denom + (size_t)N * 4;

    {
        const int total = N * 128;
        init_buffers<<<(total + 255) / 256, 256, 0, stream>>>(mbuf, denom, out, N * 4, total);
    }
    {
        dim3 grid((N + 15) / 16, 8);
        dim3 block(32, 4);
        fused_qkvs_gemm<<<grid, block, 0, stream>>>(x, Wq, bq, Wk, bk, Wv, bv,
                                                    Wsk, bsk, q, k, v, xsk, N);
    }
    {
        dim3 grid((E + 3) / 4);
        dim3 block(32, 4);
        edge_logits<<<grid, block, 0, stream>>>(edge_attr, edge_index, We, q, k,
                                                alpha, mbuf, E);
    }
    {
        dim3 grid((E + 3) / 4);
        dim3 block(32, 4);
        edge_scatter<<<grid, block, 0, stream>>>(edge_index, alpha, mbuf, v,
                                                 denom, out, E);
    }
    {
        dim3 grid((N + 3) / 4);
        dim3 block(32, 4);
        finalize<<<grid, block, 0, stream>>>(denom, xsk, Wbeta, out, N);
    }
}